// TripletLoss_40089224741570
// MI455X (gfx1250) — compile-verified
//
#include <hip/hip_runtime.h>

// Triplet loss (batch-hard mining) for X[8192,128] fp32, labels[8192] int.
// Fused fp32-WMMA Gram matrix + squared-distance hardest-pos/neg mining.
//   - V_WMMA_F32_16X16X4_F32, K=128 -> 32 chained WMMAs per 16x16 tile.
//   - B tiles staged via GLOBAL_LOAD_ASYNC_TO_LDS_B128, double-buffered,
//     synchronized with s_wait_asynccnt (CDNA5 async path).
//   - sqrt deferred out of the inner loop (monotone under max/min).
// Workspace layout (floats): sq[n] | AP[NSPLIT][n] | AN[NSPLIT][n]

typedef __attribute__((ext_vector_type(2))) float v2f;
typedef __attribute__((ext_vector_type(8))) float v8f;

#define DIMK     128
#define NSPLIT   8
#define MARGIN   0.3f
#define EPS_D2   1e-12f
#define POS_INF  __builtin_huge_valf()

// ---------------------------------------------------------------------------
// Kernel 0: sq[row] = sum_c X[row][c]^2.  One wave32 per row.
// ---------------------------------------------------------------------------
__global__ void sq_kernel(const float* __restrict__ X, float* __restrict__ sq,
                          int n) {
  const int lane = threadIdx.x & 31;
  const int wave = threadIdx.x >> 5;
  const int row  = blockIdx.x * 8 + wave;
  if (row >= n) return;
  const float4* rp = (const float4*)(X + (size_t)row * DIMK);
  float4 v = rp[lane];                       // 32 lanes x 4 floats = 128
  float s = v.x * v.x + v.y * v.y + v.z * v.z + v.w * v.w;
#pragma unroll
  for (int m = 16; m >= 1; m >>= 1) s += __shfl_xor(s, m, 32);
  if (lane == 0) sq[row] = s;
}

// ---------------------------------------------------------------------------
// Async DMA: 32 bytes (two b128) global -> LDS, tracked by ASYNCcnt.
//   ldsdst: LDS byte address (low 32 bits of flat shared pointer)
//   goff  : per-lane byte offset from SGPR base
//   NOTE: INST_OFFSET is added on both the LDS and global side (ISA 15.18.3).
// ---------------------------------------------------------------------------
__device__ __forceinline__ void async_copy_32B(unsigned ldsdst, unsigned goff,
                                               const float* base) {
  asm volatile("global_load_async_to_lds_b128 %0, %1, %2"
               :: "v"(ldsdst), "v"(goff), "s"(base) : "memory");
  asm volatile("global_load_async_to_lds_b128 %0, %1, %2 offset:16"
               :: "v"(ldsdst), "v"(goff), "s"(base) : "memory");
}

// ---------------------------------------------------------------------------
// Kernel 1: fused Gram-tile WMMA + hardest mining (on squared distances).
//   block = 256 threads = 8 waves; wave w owns rows i0 = bx*128 + 16w .. +15.
//   grid.y = NSPLIT splits the j-dimension for occupancy.
// ---------------------------------------------------------------------------
__global__ void __launch_bounds__(256)
tile_kernel(const float* __restrict__ X, const int* __restrict__ tgt,
            const float* __restrict__ sq, float* __restrict__ AP,
            float* __restrict__ AN, int n) {
  const int lane  = threadIdx.x & 31;
  const int wave  = threadIdx.x >> 5;
  const int half  = lane >> 4;          // 0: lanes 0-15, 1: lanes 16-31
  const int lrow  = lane & 15;          // M (for A) / N (for B) index
  const int i0    = blockIdx.x * 128 + wave * 16;

  // ---- Preload A tile (16 rows x 128) into WMMA 16x4 f32 fragments.
  // Lane layout: lane(M=lrow): v[0]=K0, v[1]=K1 ; lanes 16-31 carry K2,K3.
  v2f afrag[DIMK / 4];
  {
    const float* arow = X + (size_t)(i0 + lrow) * DIMK + 2 * half;
#pragma unroll
    for (int k = 0; k < DIMK / 4; ++k)
      afrag[k] = *(const v2f*)(arow + 4 * k);
  }

  // C/D layout: VGPR r holds M=r (lanes 0-15) and M=r+8 (lanes 16-31).
  float sqi[8];
  int   ti[8];
#pragma unroll
  for (int r = 0; r < 8; ++r) {
    const int irow = i0 + r + 8 * half;
    sqi[r] = sq[irow];
    ti[r]  = tgt[irow];
  }

  float ap2[8], an2[8];   // running max/min of (sqj - 2*g); sqi added later
#pragma unroll
  for (int r = 0; r < 8; ++r) { ap2[r] = -POS_INF; an2[r] = POS_INF; }

  // ---- Double-buffered LDS for the shared 16x128 B tile (stride 132).
  __shared__ float Bt[2][16 * 132];
  const int brow = threadIdx.x >> 4;         // 0..15
  const int bcol = (threadIdx.x & 15) * 8;   // 0..120 step 8
  unsigned ldsdst[2];
  ldsdst[0] = (unsigned)(size_t)&Bt[0][brow * 132 + bcol];
  ldsdst[1] = (unsigned)(size_t)&Bt[1][brow * 132 + bcol];

  const int T   = (n / 16) / NSPLIT;
  const int jt0 = blockIdx.y * T;
  const unsigned gstep = 16u * DIMK * (unsigned)sizeof(float);
  unsigned gthread =
      (unsigned)(((jt0 * 16 + brow) * DIMK + bcol) * sizeof(float));

  async_copy_32B(ldsdst[0], gthread, X);     // prefetch tile 0

  for (int t = 0; t < T; ++t) {
    if (t + 1 < T) {   // prefetch next tile, then retire current (in-order)
      async_copy_32B(ldsdst[(t + 1) & 1], gthread + (unsigned)(t + 1) * gstep,
                     X);
      asm volatile("s_wait_asynccnt 0x2" ::: "memory");
    } else {
      asm volatile("s_wait_asynccnt 0x0" ::: "memory");
    }

    const int   j0  = (jt0 + t) * 16;
    const float sqj = sq[j0 + lrow];
    const int   tj  = tgt[j0 + lrow];

    __syncthreads();                         // all waves' DMA visible

    // ---- 32 chained fp32 WMMAs: D = A(16x4) * B(4x16) + C
    v8f acc = {};
    const float* browp = &Bt[t & 1][lrow * 132 + 2 * half];
#pragma unroll
    for (int k = 0; k < DIMK / 4; ++k) {
      v2f b = *(const v2f*)(browp + 4 * k);
      acc = __builtin_amdgcn_wmma_f32_16x16x4_f32(
          /*neg_a=*/false, afrag[k], /*neg_b=*/false, b,
          /*c_mod=*/(short)0, acc, /*reuse_a=*/false, /*reuse_b=*/false);
    }

    // ---- Epilogue (squared-distance domain, sqrt deferred).
#pragma unroll
    for (int r = 0; r < 8; ++r) {
      const float val  = fmaf(-2.0f, acc[r], sqj);   // sqj - 2*g
      const bool  same = (ti[r] == tj);
      ap2[r] = fmaxf(ap2[r], same ? val : -POS_INF);
      an2[r] = fminf(an2[r], same ? POS_INF : val);
    }

    __syncthreads();   // buf[t&1] free for the prefetch issued next iteration
  }

  // ---- Reduce over the 16 N-lanes of each half-wave, finish, store.
#pragma unroll
  for (int r = 0; r < 8; ++r) {
#pragma unroll
    for (int m = 8; m >= 1; m >>= 1) {
      ap2[r] = fmaxf(ap2[r], __shfl_xor(ap2[r], m, 32));
      an2[r] = fminf(an2[r], __shfl_xor(an2[r], m, 32));
    }
  }
  if (lrow == 0) {
    const size_t base = (size_t)blockIdx.y * n;
#pragma unroll
    for (int r = 0; r < 8; ++r) {
      const int irow = i0 + r + 8 * half;
      AP[base + irow] = sqrtf(fmaxf(sqi[r] + ap2[r], EPS_D2));
      AN[base + irow] = sqrtf(fmaxf(sqi[r] + an2[r], EPS_D2));
    }
  }
}

// ---------------------------------------------------------------------------
// Kernel 2: combine split partials, margin-ranking loss, mean. Single block.
// ---------------------------------------------------------------------------
__global__ void __launch_bounds__(256)
loss_kernel(const float* __restrict__ AP, const float* __restrict__ AN,
            float* __restrict__ out, int n) {
  float sum = 0.0f;
  for (int row = threadIdx.x; row < n; row += 256) {
    float ap = -POS_INF, an = POS_INF;
#pragma unroll
    for (int s = 0; s < NSPLIT; ++s) {
      ap = fmaxf(ap, AP[(size_t)s * n + row]);
      an = fminf(an, AN[(size_t)s * n + row]);
    }
    sum += fmaxf(0.0f, MARGIN + ap - an);
  }
#pragma unroll
  for (int m = 16; m >= 1; m >>= 1) sum += __shfl_xor(sum, m, 32);

  __shared__ float wsum[8];
  const int lane = threadIdx.x & 31, wave = threadIdx.x >> 5;
  if (lane == 0) wsum[wave] = sum;
  __syncthreads();
  if (threadIdx.x == 0) {
    float tot = 0.0f;
#pragma unroll
    for (int w = 0; w < 8; ++w) tot += wsum[w];
    out[0] = tot / (float)n;
  }
}

// ---------------------------------------------------------------------------
extern "C" void kernel_launch(void* const* d_in, const int* in_sizes, int n_in,
                              void* d_out, int out_size, void* d_ws,
                              size_t ws_size, hipStream_t stream) {
  const float* X   = (const float*)d_in[0];
  const int*   tgt = (const int*)d_in[1];
  const int n = in_sizes[1];                 // 8192 rows (DIMK = 128 fixed)

  float* ws = (float*)d_ws;
  float* sq = ws;                            // n floats
  float* AP = ws + n;                        // NSPLIT * n floats
  float* AN = AP + (size_t)NSPLIT * n;       // NSPLIT * n floats

  sq_kernel<<<n / 8, 256, 0, stream>>>(X, sq, n);
  dim3 grid(n / 128, NSPLIT);
  tile_kernel<<<grid, 256, 0, stream>>>(X, tgt, sq, AP, AN, n);
  loss_kernel<<<1, 256, 0, stream>>>(AP, AN, (float*)d_out, n);
}